// MyConv2d_46299747450914
// MI455X (gfx1250) — compile-verified
//
#include <hip/hip_runtime.h>

typedef __attribute__((ext_vector_type(2))) float v2f;
typedef __attribute__((ext_vector_type(8))) float v8f;

namespace {
constexpr int N_IMG    = 32;
constexpr int C_IN     = 128;
constexpr int HW       = 56;
constexpr int K_OUT    = 256;
constexpr int KDIM     = C_IN * 9;          // 1152 (im2col K)
constexpr int PX_TOTAL = N_IMG * HW * HW;   // 100352 (GEMM N)
constexpr int TILE_OC  = 64;
constexpr int TILE_PX  = 64;
constexpr int TILE_K   = 16;
constexpr int NCHUNK   = KDIM / TILE_K;     // 72
constexpr int A_STR    = TILE_K + 2;        // 18: conflict-free b64 fragment reads
constexpr int B_STR    = TILE_K + 2;        // 18: B stored transposed [px][k]
}

// Implicit-GEMM conv3x3 using V_WMMA_F32_16X16X4_F32 (full f32 precision).
// Block: 128 threads = 4 waves in a 2x2 (oc x px) arrangement.
// Block tile: 64 oc x 64 px; wave tile: 32 x 32 = 2x2 WMMA accumulators.
// Double-buffered LDS: one barrier/iter, next chunk's global loads overlap WMMAs.
__global__ __launch_bounds__(128, 2)
void conv3x3_wmma_f32(const float* __restrict__ x,
                      const float* __restrict__ w,
                      float* __restrict__ out) {
  __shared__ float aLds[2][TILE_OC * A_STR];   // [buf][oc][k]
  __shared__ float bLds[2][TILE_PX * B_STR];   // [buf][px][k]  (transposed!)

  const int tid    = threadIdx.x;
  const int lane   = tid & 31;
  const int wave   = tid >> 5;
  const int waveOc = (wave >> 1) * 32;
  const int wavePx = (wave & 1) * 32;

  const int blockOc = blockIdx.y * TILE_OC;
  const int blockPx = blockIdx.x * TILE_PX;

  v8f acc[2][2] = {};

  // A staging: weights OIHW == [oc][k] row-major (k = c*9+r*3+s) -> contiguous.
  // Thread -> (oc = tid/2, k0 = (tid&1)*8): 8 consecutive floats (2x b128).
  const int aOc = tid >> 1;
  const int aK0 = (tid & 1) * 8;
  const float* wRow = w + (size_t)(blockOc + aOc) * KDIM + aK0;

  // B staging (im2col gather): kRow = tid>>3 (0..15), px = (tid&7)*8 + j.
  const int bK   = tid >> 3;
  const int bPx0 = (tid & 7) * 8;

  float aReg[8];
  float bReg[8];

  // Global -> registers for chunk starting at k0.
  auto stageLoad = [&](int k0) {
    #pragma unroll
    for (int j = 0; j < 8; ++j) aReg[j] = wRow[k0 + j];
    const int k  = k0 + bK;
    const int c  = k / 9;
    const int rs = k % 9;
    const int r  = rs / 3;
    const int s  = rs % 3;
    #pragma unroll
    for (int j = 0; j < 8; ++j) {
      const int px  = blockPx + bPx0 + j;
      const int n   = px / (HW * HW);
      const int rem = px % (HW * HW);
      const int oh  = rem / HW;
      const int ow  = rem % HW;
      const int ih  = oh + r - 1;
      const int iw  = ow + s - 1;
      float v = 0.0f;
      if ((unsigned)ih < (unsigned)HW && (unsigned)iw < (unsigned)HW)
        v = x[(((size_t)n * C_IN + c) * HW + ih) * HW + iw];
      bReg[j] = v;
    }
  };

  // Registers -> LDS buffer `buf`. B goes in transposed: [px][k].
  auto stageStore = [&](int buf) {
    #pragma unroll
    for (int j = 0; j < 8; ++j)
      aLds[buf][aOc * A_STR + aK0 + j] = aReg[j];
    #pragma unroll
    for (int j = 0; j < 8; ++j)
      bLds[buf][(bPx0 + j) * B_STR + bK] = bReg[j];
  };

  // Prologue: stage chunk 0.
  stageLoad(0);
  stageStore(0);
  __syncthreads();

  const int m  = lane & 15;          // A row / B col within 16
  const int kp = (lane >> 4) * 2;    // K-pair selector per ISA A/B layout

  for (int ch = 0; ch < NCHUNK; ++ch) {
    const int  cur  = ch & 1;
    const bool more = (ch + 1) < NCHUNK;

    // Issue next chunk's global loads early: latency hides under the WMMAs.
    if (more) stageLoad((ch + 1) * TILE_K);

    // Compute: 4 sub-steps of K=4, 16 WMMAs per chunk per wave.
    #pragma unroll
    for (int kk = 0; kk < 4; ++kk) {
      v2f aF[2], bF[2];
      #pragma unroll
      for (int t = 0; t < 2; ++t)
        aF[t] = *(const v2f*)&aLds[cur][(waveOc + t * 16 + m) * A_STR + kk * 4 + kp];
      #pragma unroll
      for (int t = 0; t < 2; ++t)
        bF[t] = *(const v2f*)&bLds[cur][(wavePx + t * 16 + m) * B_STR + kk * 4 + kp];
      #pragma unroll
      for (int to = 0; to < 2; ++to)
        #pragma unroll
        for (int tp = 0; tp < 2; ++tp)
          acc[to][tp] = __builtin_amdgcn_wmma_f32_16x16x4_f32(
              false, aF[to], false, bF[tp], (short)0, acc[to][tp],
              false, false);
    }

    // Publish next chunk into the alternate buffer; single barrier per iter.
    if (more) stageStore(cur ^ 1);
    __syncthreads();
  }

  // Epilogue: D layout VGPR v -> row (lane>>4)*8 + v, col lane&15. Output NKHW.
  #pragma unroll
  for (int tp = 0; tp < 2; ++tp) {
    const int px  = blockPx + wavePx + tp * 16 + (lane & 15);
    const int n   = px / (HW * HW);
    const int rem = px % (HW * HW);
    #pragma unroll
    for (int to = 0; to < 2; ++to) {
      #pragma unroll
      for (int v = 0; v < 8; ++v) {
        const int oc = blockOc + waveOc + to * 16 + (lane >> 4) * 8 + v;
        out[((size_t)n * K_OUT + oc) * (HW * HW) + rem] = acc[to][tp][v];
      }
    }
  }
}

extern "C" void kernel_launch(void* const* d_in, const int* in_sizes, int n_in,
                              void* d_out, int out_size, void* d_ws, size_t ws_size,
                              hipStream_t stream) {
  const float* x = (const float*)d_in[0];   // (32,128,56,56) f32
  const float* w = (const float*)d_in[1];   // (256,128,3,3)  f32
  float* out = (float*)d_out;               // (32,256,56,56) f32
  dim3 grid(PX_TOTAL / TILE_PX, K_OUT / TILE_OC, 1);  // (1568, 4)
  conv3x3_wmma_f32<<<grid, 128, 0, stream>>>(x, w, out);
}